// TokenTopKRouter_79611513799333
// MI455X (gfx1250) — compile-verified
//
#include <hip/hip_runtime.h>
#include <hip/hip_bf16.h>

// ---- problem constants (fixed by the reference) ----
#define TOKENS   16384      // B*S = 4*4096
#define HDIM     2048
#define NEXP     64
#define TOPK     8
#define TOK_BLK  16         // tokens per block
#define KC       128        // K chunk staged in LDS
#define NCHUNK   (HDIM / KC)
#define LDA      132        // padded LDS row stride in floats (bank step = 4 dwords)
#define PROBS_LD (NEXP + 4)

#define AS1 __attribute__((address_space(1)))
#define AS3 __attribute__((address_space(3)))

#if __has_builtin(__builtin_amdgcn_global_load_async_to_lds_b128) && \
    __has_builtin(__builtin_amdgcn_s_wait_asynccnt)
#define USE_ASYNC 1
#else
#define USE_ASYNC 0
#endif

typedef __attribute__((ext_vector_type(2))) float v2f;
typedef __attribute__((ext_vector_type(8))) float v8f;
typedef __attribute__((ext_vector_type(4))) int   v4i;

__global__ void router_init_ws(float* ws) {
    if (threadIdx.x < 2 * NEXP) ws[threadIdx.x] = 0.0f;
}

__global__ __launch_bounds__(128) void router_main(const float* __restrict__ x,
                                                   const float* __restrict__ W,
                                                   float* __restrict__ out,
                                                   float* __restrict__ gws) {
    __shared__ float ldsA[2][TOK_BLK * LDA];   // double-buffered 16x128 x-tile
    __shared__ float ldsW[2][NEXP * LDA];      // double-buffered 64x128 W-tile
    __shared__ float ldsP[TOK_BLK * PROBS_LD]; // logits -> probs
    __shared__ float smLoad[NEXP];             // per-expert selection counts

    const int tid  = threadIdx.x;
    const int wave = tid >> 5;       // 0..3, expert group
    const int lane = tid & 31;
    const int half = lane >> 4;      // 0 or 1 (K sub-pair select)
    const int l15  = lane & 15;      // row within 16
    const int tokBase = blockIdx.x * TOK_BLK;

    if (tid < NEXP) smLoad[tid] = 0.0f;

    // staging assignments (coalesced b128 streams; no VGPR staging)
    const int rowA = tid >> 3;            // 0..15
    const int colA = (tid & 7) * 16;      // 0..112
    const int rowW = tid >> 1;            // 0..63
    const int colW = (tid & 1) * 64;      // 0 or 64
    const float* __restrict__ srcA = x + (size_t)(tokBase + rowA) * HDIM + colA;
    const float* __restrict__ srcW = W + (size_t)rowW * HDIM + colW;

    v8f acc = {0.f, 0.f, 0.f, 0.f, 0.f, 0.f, 0.f, 0.f};

#if USE_ASYNC
    // ---- async global->LDS staging (ASYNCcnt), no registers involved ----
    #define STAGE(buf, k0)                                                          \
        do {                                                                        \
            _Pragma("unroll")                                                       \
            for (int i = 0; i < 4; ++i)                                             \
                __builtin_amdgcn_global_load_async_to_lds_b128(                     \
                    (AS1 v4i*)(srcA + (k0) + 4 * i),                                \
                    (AS3 v4i*)(&ldsA[buf][rowA * LDA + colA + 4 * i]), 0, 0);       \
            _Pragma("unroll")                                                       \
            for (int i = 0; i < 16; ++i)                                            \
                __builtin_amdgcn_global_load_async_to_lds_b128(                     \
                    (AS1 v4i*)(srcW + (k0) + 4 * i),                                \
                    (AS3 v4i*)(&ldsW[buf][rowW * LDA + colW + 4 * i]), 0, 0);       \
        } while (0)

    STAGE(0, 0);
    for (int c = 0; c < NCHUNK; ++c) {
        __builtin_amdgcn_s_wait_asynccnt(0);  // chunk c landed in buf[c&1]
        __syncthreads();                      // visible to all; prior reads of other buf done
        if (c + 1 < NCHUNK) STAGE((c + 1) & 1, (c + 1) * KC);  // overlaps WMMAs

        const float* aRow = &ldsA[c & 1][l15 * LDA + 2 * half];
        const float* bRow = &ldsW[c & 1][(wave * 16 + l15) * LDA + 2 * half];
        #pragma unroll 8
        for (int kk = 0; kk < KC; kk += 4) {
            v2f a = *(const v2f*)(aRow + kk);
            v2f b = *(const v2f*)(bRow + kk);
            acc = __builtin_amdgcn_wmma_f32_16x16x4_f32(
                false, a, false, b, (short)0, acc, false, false);
        }
    }
    #undef STAGE
#else
    // ---- fallback: simple load->store staging, two barriers per chunk ----
    for (int c = 0; c < NCHUNK; ++c) {
        const int k0 = c * KC;
        __syncthreads();  // prior chunk's reads complete before overwrite
        #pragma unroll
        for (int i = 0; i < 4; ++i) {
            float4 t = *(const float4*)(srcA + k0 + 4 * i);
            *(float4*)(&ldsA[c & 1][rowA * LDA + colA + 4 * i]) = t;
        }
        #pragma unroll
        for (int i = 0; i < 16; ++i) {
            float4 t = *(const float4*)(srcW + k0 + 4 * i);
            *(float4*)(&ldsW[c & 1][rowW * LDA + colW + 4 * i]) = t;
        }
        __syncthreads();

        const float* aRow = &ldsA[c & 1][l15 * LDA + 2 * half];
        const float* bRow = &ldsW[c & 1][(wave * 16 + l15) * LDA + 2 * half];
        #pragma unroll 8
        for (int kk = 0; kk < KC; kk += 4) {
            v2f a = *(const v2f*)(aRow + kk);
            v2f b = *(const v2f*)(bRow + kk);
            acc = __builtin_amdgcn_wmma_f32_16x16x4_f32(
                false, a, false, b, (short)0, acc, false, false);
        }
    }
#endif

    // ---- spill logits tile to LDS per documented C/D layout ----
    // VGPR r: lanes 0-15 -> (M=r, N=lane); lanes 16-31 -> (M=r+8, N=lane-16)
    #pragma unroll
    for (int r = 0; r < 8; ++r) {
        const int m = r + 8 * half;
        const int n = l15 + wave * 16;
        ldsP[m * PROBS_LD + n] = acc[r];
    }
    __syncthreads();

    // ---- softmax + top-8: 8 lanes per token, lane owns experts [8j, 8j+8) ----
    {
        const int grp = tid >> 3;   // token within block
        const int lig = tid & 7;    // lane in 8-lane group
        float* rowP = &ldsP[grp * PROBS_LD + lig * 8];

        float ev[8];
        #pragma unroll
        for (int q = 0; q < 8; ++q) ev[q] = rowP[q];

        // group max
        float mx = ev[0];
        #pragma unroll
        for (int q = 1; q < 8; ++q) mx = fmaxf(mx, ev[q]);
        #pragma unroll
        for (int m = 1; m < 8; m <<= 1) mx = fmaxf(mx, __shfl_xor(mx, m, 32));

        // exp + group sum
        float sum = 0.0f;
        #pragma unroll
        for (int q = 0; q < 8; ++q) { ev[q] = __expf(ev[q] - mx); sum += ev[q]; }
        #pragma unroll
        for (int m = 1; m < 8; m <<= 1) sum += __shfl_xor(sum, m, 32);
        const float inv = 1.0f / sum;

        // write probs back (importance column sums read these)
        #pragma unroll
        for (int q = 0; q < 8; ++q) rowP[q] = ev[q] * inv;

        // 8 rounds of cross-lane argmax (lowest-index tie-break, like lax.top_k)
        float tsum = 0.0f, myVal = 0.0f;
        int myIdx = 0;
        #pragma unroll
        for (int t = 0; t < 8; ++t) {
            float bv = -1.0f; int bi = 0;
            #pragma unroll
            for (int q = 0; q < 8; ++q)
                if (ev[q] > bv) { bv = ev[q]; bi = lig * 8 + q; }
            #pragma unroll
            for (int m = 1; m < 8; m <<= 1) {
                float ov = __shfl_xor(bv, m, 32);
                int   oi = __shfl_xor(bi, m, 32);
                if (ov > bv || (ov == bv && oi < bi)) { bv = ov; bi = oi; }
            }
            if ((bi >> 3) == lig) {        // this lane owns the winner
                #pragma unroll
                for (int q = 0; q < 8; ++q)
                    if ((bi & 7) == q) ev[q] = -1.0f;   // consume
                atomicAdd(&smLoad[bi], 1.0f);           // ds_add_f32
            }
            tsum += bv;
            if (lig == t) { myVal = bv; myIdx = bi; }
        }

        const float denom  = fmaxf(tsum * inv, 1e-8f);  // clip like reference
        const float wscale = inv / denom;               // w = (exp*inv)/denom
        const size_t token = (size_t)tokBase + grp;
        out[token * TOPK + lig] = (float)myIdx;
        out[(size_t)TOKENS * TOPK + token * TOPK + lig] = myVal * wscale;
    }
    __syncthreads();

    // ---- per-expert partial sums -> global workspace ----
    if (tid < NEXP) {
        float s = 0.0f;
        #pragma unroll
        for (int m = 0; m < TOK_BLK; ++m) s += ldsP[m * PROBS_LD + tid];
        atomicAdd(&gws[tid], s);                  // importance partial
        atomicAdd(&gws[NEXP + tid], smLoad[tid]); // load partial
    }
}

__global__ void router_aux(const float* __restrict__ gws, float* __restrict__ out) {
    __shared__ float red[NEXP];
    const int e = threadIdx.x;
    const float imp  = gws[e] * (1.0f / (float)TOKENS);
    const float load = gws[NEXP + e] * (1.0f / ((float)TOKENS * (float)TOPK));
    red[e] = imp * load;
    __syncthreads();
    if (e == 0) {
        float s = 0.0f;
        for (int i = 0; i < NEXP; ++i) s += red[i];
        out[(size_t)2 * TOKENS * TOPK] = (float)NEXP * s;
    }
}

extern "C" void kernel_launch(void* const* d_in, const int* in_sizes, int n_in,
                              void* d_out, int out_size, void* d_ws, size_t ws_size,
                              hipStream_t stream) {
    const float* x = (const float*)d_in[0];   // [4,4096,2048] f32
    const float* W = (const float*)d_in[1];   // [64,2048] f32
    float* out = (float*)d_out;               // [idx 131072][w 131072][aux 1]
    float* ws  = (float*)d_ws;                // [imp 64][load 64]

    router_init_ws<<<1, 128, 0, stream>>>(ws);
    router_main<<<TOKENS / TOK_BLK, 128, 0, stream>>>(x, W, out, ws);
    router_aux<<<1, NEXP, 0, stream>>>(ws, out);
}